// ScatterMLP_23021024707051
// MI455X (gfx1250) — compile-verified
//
#include <hip/hip_runtime.h>
#include <hip/hip_bf16.h>

typedef unsigned long long u64;

#define CCH 128          // channels
#define RPB 8            // rows (waves) per block
#define CHUNK 32         // edges staged per async chunk

// ---------------- CSR build ----------------

__global__ void hist_kernel(const int* __restrict__ row, int* __restrict__ deg, int E) {
    int e = blockIdx.x * blockDim.x + threadIdx.x;
    if (e < E) atomicAdd(&deg[row[e]], 1);
}

__global__ void scan_block(const int* __restrict__ deg, int* __restrict__ excl,
                           int* __restrict__ bsums, int n) {
    __shared__ int tmp[256];
    int i = blockIdx.x * 256 + threadIdx.x;
    int v = (i < n) ? deg[i] : 0;
    tmp[threadIdx.x] = v;
    __syncthreads();
    // Hillis-Steele inclusive scan
    for (int off = 1; off < 256; off <<= 1) {
        int t = (threadIdx.x >= off) ? tmp[threadIdx.x - off] : 0;
        __syncthreads();
        tmp[threadIdx.x] += t;
        __syncthreads();
    }
    if (i < n) excl[i] = tmp[threadIdx.x] - v;          // exclusive within block
    if (threadIdx.x == 255) bsums[blockIdx.x] = tmp[255];
}

__global__ void scan_sums(int* __restrict__ bsums, int nb) {
    if (blockIdx.x == 0 && threadIdx.x == 0) {
        int acc = 0;
        for (int b = 0; b < nb; ++b) { int t = bsums[b]; bsums[b] = acc; acc += t; }
    }
}

__global__ void scan_finalize(int* __restrict__ rowptr, int* __restrict__ wofs,
                              const int* __restrict__ bsums, int n, int E) {
    int i = blockIdx.x * 256 + threadIdx.x;
    if (i < n) {
        int r = rowptr[i] + bsums[blockIdx.x];
        rowptr[i] = r;
        wofs[i]   = r;
    }
    if (i == 0) rowptr[n] = E;
}

__global__ void scatter_kernel(const int* __restrict__ row, const int* __restrict__ col,
                               const float* __restrict__ vals, int* __restrict__ wofs,
                               u64* __restrict__ epack, int E) {
    int e = blockIdx.x * blockDim.x + threadIdx.x;
    if (e < E) {
        int p = atomicAdd(&wofs[row[e]], 1);
        epack[p] = ((u64)__float_as_uint(vals[e]) << 32) | (unsigned)col[e];
    }
}

// ---------------- init / snapshot ----------------

// buf0 <- x ; out slot0 <- x  (S0 raw; becomes S0-S1 later)
__global__ void init_kernel(const float* __restrict__ x, float* __restrict__ buf0,
                            float* __restrict__ out, size_t nc) {
    size_t i = (size_t)blockIdx.x * blockDim.x + threadIdx.x;
    if (i < nc) {
        float v = x[i];
        buf0[i] = v;
        out[i * 6 + 0] = v;
    }
}

// new snapshot S_j in cur: out[..,j-1] = S_{j-1} - S_j ; out[..,j] = S_j
__global__ void snap_kernel(const float* __restrict__ cur, float* __restrict__ out,
                            int j, size_t nc) {
    size_t i = (size_t)blockIdx.x * blockDim.x + threadIdx.x;
    if (i < nc) {
        float v = cur[i];
        size_t o = i * 6;
        out[o + j - 1] = out[o + j - 1] - v;
        out[o + j]     = v;
    }
}

// ---------------- gather SpMM (no atomics, L2-resident x) ----------------
// One wave32 per row; lane owns 4 channels (float4 -> b128, 512B coalesced/edge).
// CSR edge segment is staged into LDS in 32-edge chunks via async global->LDS
// (ASYNCcnt), double-buffered so the next chunk is in flight during the FMAs.
// 4 independent accumulators keep 4 row-gathers outstanding per wave (MLP).

__device__ __forceinline__ void edge_fma(const u64* __restrict__ buf, int k,
                                         const float* __restrict__ xin, int lane,
                                         float4& a) {
    u64 p = buf[k];                              // uniform ds_load_b64 broadcast
    int   c = (int)(unsigned)p;
    float v = __uint_as_float((unsigned)(p >> 32));
    const float4 xv = *(const float4*)(xin + ((size_t)c << 7) + ((size_t)lane << 2));
    a.x = fmaf(v, xv.x, a.x);
    a.y = fmaf(v, xv.y, a.y);
    a.z = fmaf(v, xv.z, a.z);
    a.w = fmaf(v, xv.w, a.w);
}

__global__ __launch_bounds__(256) void spmm_kernel(const int* __restrict__ rowptr,
                                                   const u64* __restrict__ epack,
                                                   const float* __restrict__ xin,
                                                   float* __restrict__ yout,
                                                   int nrows) {
    __shared__ u64 sh[RPB][2][CHUNK];
    const int wave = threadIdx.x >> 5;
    const int lane = threadIdx.x & 31;
    const int r = blockIdx.x * RPB + wave;
    if (r >= nrows) return;

    int start = __builtin_amdgcn_readfirstlane(rowptr[r]);
    int end   = __builtin_amdgcn_readfirstlane(rowptr[r + 1]);

    // LDS byte offsets for this lane's slot in each buffer
    const unsigned lds0 = (unsigned)(size_t)&sh[wave][0][lane];
    const unsigned lds1 = (unsigned)(size_t)&sh[wave][1][lane];

    float4 a0 = make_float4(0.f, 0.f, 0.f, 0.f);
    float4 a1 = a0, a2 = a0, a3 = a0;
    const int nch = (end - start + CHUNK - 1) >> 5;

    auto issue = [&](int ci) {
        int e = start + (ci << 5) + lane;
        if (e < end) {
            unsigned loff = (ci & 1) ? lds1 : lds0;
            unsigned voff = (unsigned)e * 8u;   // byte offset from epack base
            asm volatile("global_load_async_to_lds_b64 %0, %1, %2 offset:0"
                         :: "v"(loff), "v"(voff), "s"(epack)
                         : "memory");
        }
    };

    if (nch > 0) issue(0);
    for (int ci = 0; ci < nch; ++ci) {
        if (ci + 1 < nch) {
            issue(ci + 1);
            asm volatile("s_wait_asynccnt 0x1" ::: "memory"); // chunk ci landed
        } else {
            asm volatile("s_wait_asynccnt 0x0" ::: "memory");
        }
        const u64* buf = &sh[wave][ci & 1][0];
        const int n = min(CHUNK, end - start - (ci << 5));
        int k = 0;
        for (; k + 4 <= n; k += 4) {            // 4 independent gathers in flight
            edge_fma(buf, k + 0, xin, lane, a0);
            edge_fma(buf, k + 1, xin, lane, a1);
            edge_fma(buf, k + 2, xin, lane, a2);
            edge_fma(buf, k + 3, xin, lane, a3);
        }
        for (; k < n; ++k) edge_fma(buf, k, xin, lane, a0);
    }
    float4 acc;
    acc.x = (a0.x + a1.x) + (a2.x + a3.x);
    acc.y = (a0.y + a1.y) + (a2.y + a3.y);
    acc.z = (a0.z + a1.z) + (a2.z + a3.z);
    acc.w = (a0.w + a1.w) + (a2.w + a3.w);
    *(float4*)(yout + ((size_t)r << 7) + ((size_t)lane << 2)) = acc;
}

// ---------------- launch ----------------

extern "C" void kernel_launch(void* const* d_in, const int* in_sizes, int n_in,
                              void* d_out, int out_size, void* d_ws, size_t ws_size,
                              hipStream_t stream) {
    const float* x    = (const float*)d_in[0];
    const float* vals = (const float*)d_in[1];
    const int*   row  = (const int*)d_in[2];
    const int*   col  = (const int*)d_in[3];
    float* out = (float*)d_out;

    const int E = in_sizes[1];
    const int N = in_sizes[0] / CCH;
    const size_t NC = (size_t)N * CCH;

    // workspace layout
    char* w = (char*)d_ws;
    float* buf0   = (float*)w;                 // NC floats
    float* buf1   = buf0 + NC;                 // NC floats
    u64*   epack  = (u64*)(buf1 + NC);         // E packed (col,val)
    int*   rowptr = (int*)(epack + E);         // N+1
    int*   wofs   = rowptr + (N + 1);          // N
    int*   deg    = wofs + N;                  // N
    int*   bsums  = deg + N;                   // nblkN

    const int nblkE  = (E + 255) / 256;
    const int nblkN  = (N + 255) / 256;
    const int nblkNC = (int)((NC + 255) / 256);
    const int nblkR  = (N + RPB - 1) / RPB;

    // ---- CSR build ----
    hipMemsetAsync(deg, 0, sizeof(int) * N, stream);
    hist_kernel<<<nblkE, 256, 0, stream>>>(row, deg, E);
    scan_block<<<nblkN, 256, 0, stream>>>(deg, rowptr, bsums, N);
    scan_sums<<<1, 1, 0, stream>>>(bsums, nblkN);
    scan_finalize<<<nblkN, 256, 0, stream>>>(rowptr, wofs, bsums, N, E);
    scatter_kernel<<<nblkE, 256, 0, stream>>>(row, col, vals, wofs, epack, E);

    // ---- S0 ----
    init_kernel<<<nblkNC, 256, 0, stream>>>(x, buf0, out, NC);

    // ---- 16 diffusion steps, snapshots at t = 1,2,4,8,16 ----
    float* cur = buf0;
    float* nxt = buf1;
    int j = 0, nextpow = 1;
    for (int t = 1; t <= 16; ++t) {
        spmm_kernel<<<nblkR, 256, 0, stream>>>(rowptr, epack, cur, nxt, N);
        float* tmp = cur; cur = nxt; nxt = tmp;
        if (t == nextpow) {
            ++j;
            snap_kernel<<<nblkNC, 256, 0, stream>>>(cur, out, j, NC);
            nextpow <<= 1;
        }
    }
}